// GNNCritic_1752346657364
// MI455X (gfx1250) — compile-verified
//
#include <hip/hip_runtime.h>
#include <hip/hip_bf16.h>

// ---------------------------------------------------------------------------
// Problem constants (from reference)
// ---------------------------------------------------------------------------
#define B_   256
#define N_   100
#define C_   256
#define H_   256
#define E_   1000
#define EG_  256000
#define NTOT (B_ * N_)   // 25600

typedef __attribute__((ext_vector_type(2))) float v2f;
typedef __attribute__((ext_vector_type(8))) float v8f;

typedef unsigned int u32x4 __attribute__((ext_vector_type(4)));
typedef int          i32x4 __attribute__((ext_vector_type(4)));
typedef int          i32x8 __attribute__((ext_vector_type(8)));

#ifndef __has_builtin
#define __has_builtin(x) 0
#endif
#if __has_builtin(__builtin_amdgcn_tensor_load_to_lds) && \
    __has_builtin(__builtin_amdgcn_s_wait_tensorcnt)
#define USE_TDM 1
#else
#define USE_TDM 0
#endif

// LDS row stride used everywhere (264 = 256 + 8 pad DWORDs, matches TDM pad)
#define LDAP 264

#if USE_TDM
// ---------------------------------------------------------------------------
// Tensor Data Mover: build D# groups per CDNA5 ISA Ch.8 and issue the DMA.
//   - data_size = 2 (4 bytes)
//   - pad_enable=1, pad_interval=7 (256 DW), pad_amount=7 (8 DW) -> LDS row
//     stride of 264 DWORDs (bank-conflict padding applied by the TDM itself)
//   - gather=true: groups 2/3 carry 16 packed 16-bit row indices,
//     tile_dim1 = #valid indices
// ---------------------------------------------------------------------------
static __device__ __forceinline__ void tdm_load_to_lds(
    unsigned lds_off, unsigned long long gaddr,
    unsigned td0, unsigned td1, unsigned tile0, unsigned tile1,
    unsigned long long stride0, bool gather, i32x4 g2, i32x4 g3)
{
    u32x4 g0;
    g0[0] = 1u | (gather ? (1u << 31) : 0u);           // count=1, gather_mode, 16b idx
    g0[1] = lds_off;                                   // LDS byte address
    g0[2] = (unsigned)(gaddr & 0xffffffffull);         // global_addr[31:0]
    g0[3] = (unsigned)((gaddr >> 32) & 0x1ffffffull)   // global_addr[56:32]
          | (2u << 30);                                // type = 2 ("image")
    i32x8 g1;
    g1[0] = (int)((2u << 16)            // data_size = 4B
          | (1u << 20)                  // pad_enable
          | (7u << 22)                  // pad_interval: 256 DWORDs
          | (7u << 25));                // pad_amount:   8 DWORDs
    g1[1] = (int)((td0 & 0xffffu) << 16);              // tensor_dim0[15:0]
    g1[2] = (int)((td0 >> 16) | ((td1 & 0xffffu) << 16));
    g1[3] = (int)((td1 >> 16) | ((tile0 & 0xffffu) << 16));
    g1[4] = (int)(tile1 & 0xffffu);                    // tile_dim1, tile_dim2=0
    g1[5] = (int)(stride0 & 0xffffffffull);            // tensor_dim0_stride[31:0]
    g1[6] = (int)((stride0 >> 32) & 0xffffull);        // stride0[47:32], stride1=0
    g1[7] = 0;
#if __clang_major__ >= 23
    i32x8 g4 = {0,0,0,0,0,0,0,0};
    __builtin_amdgcn_tensor_load_to_lds(g0, g1, g2, g3, g4, 0);
#else
    __builtin_amdgcn_tensor_load_to_lds(g0, g1, g2, g3, 0);
#endif
}

static __device__ __forceinline__ unsigned lds_off_of(const void* p) {
    return (unsigned)(uintptr_t)p;   // low 32 bits of LDS aperture addr = offset
}
#endif // USE_TDM

// ---------------------------------------------------------------------------
// Full-precision f32 WMMA: D(16x16) = A(16x4) * B(4x16) + C
// ---------------------------------------------------------------------------
static __device__ __forceinline__ v8f wmma4(v2f a, v2f b, v8f c) {
    return __builtin_amdgcn_wmma_f32_16x16x4_f32(
        /*neg_a=*/false, a, /*neg_b=*/false, b,
        /*c_mod=*/(short)0, c, /*reuse_a=*/false, /*reuse_b=*/false);
}

// Per-wave 16x16 tile GEMM over K, A from LDS (row-major, stride lda),
// B from global (row-major K x ldb, column offset n0).
// Lane l: m = l&15, k-offset = 2*(l>>4); accumulator layout per ISA 7.12.2.
static __device__ __forceinline__ v8f wmma_tile_f32(
    const float* __restrict__ ldsA, int lda,
    const float* __restrict__ Wg, int ldb, int n0, int K, v8f acc)
{
    const int lane  = threadIdx.x & 31;
    const int m     = lane & 15;
    const int koff  = (lane >> 4) << 1;           // 0 or 2
    const int n     = n0 + (lane & 15);
    const float* arow = ldsA + m * lda + koff;
    const float* bcol = Wg + (size_t)koff * ldb + n;
    #pragma unroll 4
    for (int kb = 0; kb < K; kb += 4) {
        v2f a; a.x = arow[kb];               a.y = arow[kb + 1];
        v2f b; b.x = bcol[(size_t)kb * ldb]; b.y = bcol[(size_t)(kb + 1) * ldb];
        acc = wmma4(a, b, acc);
    }
    return acc;
}

// ---------------------------------------------------------------------------
// Kernel 1: xw = state @ W_gcn   (25600x256 @ 256x256)
// A-strip staged to LDS by one TDM per workgroup (wave 0 issues).
// ---------------------------------------------------------------------------
__global__ void __launch_bounds__(256)
gemm_xw_kernel(const float* __restrict__ state, const float* __restrict__ Wg,
               float* __restrict__ xw)
{
    __shared__ float A[16 * LDAP];
    const int row0 = blockIdx.x * 16;

#if USE_TDM
    if (threadIdx.x < 32) {
        i32x4 z4 = {0, 0, 0, 0};
        const unsigned long long ga =
            (unsigned long long)(uintptr_t)(state + (size_t)row0 * C_);
        tdm_load_to_lds(lds_off_of(A), ga,
                        /*td0=*/C_, /*td1=*/NTOT, /*tile0=*/C_, /*tile1=*/16,
                        /*stride0=*/C_, /*gather=*/false, z4, z4);
        __builtin_amdgcn_s_wait_tensorcnt(0);
    }
#else
    for (int idx = threadIdx.x; idx < 16 * 64; idx += blockDim.x) {
        const int r  = idx >> 6;
        const int c4 = (idx & 63) * 4;
        const float4 v = *reinterpret_cast<const float4*>(
            state + (size_t)(row0 + r) * C_ + c4);
        *reinterpret_cast<float4*>(&A[r * LDAP + c4]) = v;
    }
#endif
    __syncthreads();

    const int wave = threadIdx.x >> 5;
    const int lane = threadIdx.x & 31;
    const int mhi  = (lane >> 4) * 8;
    #pragma unroll
    for (int t = 0; t < 2; ++t) {
        const int n0 = (wave * 2 + t) * 16;
        v8f acc = {0.f,0.f,0.f,0.f,0.f,0.f,0.f,0.f};
        acc = wmma_tile_f32(A, LDAP, Wg, C_, n0, C_, acc);
        const int n = n0 + (lane & 15);
        #pragma unroll
        for (int v = 0; v < 8; ++v)
            xw[(size_t)(row0 + mhi + v) * C_ + n] = acc[v];
    }
}

// ---------------------------------------------------------------------------
// Degree / normalization kernels
// ---------------------------------------------------------------------------
__global__ void deg_init_kernel(float* __restrict__ deg) {
    const int i = blockIdx.x * blockDim.x + threadIdx.x;
    if (i < NTOT) deg[i] = 1.0f;   // self loop contributes 1
}

__global__ void deg_count_kernel(const int* __restrict__ dst, float* __restrict__ deg) {
    const int i = blockIdx.x * blockDim.x + threadIdx.x;
    if (i < EG_) atomicAdd(&deg[dst[i]], 1.0f);
}

__global__ void deg_rsqrt_kernel(const float* __restrict__ deg, float* __restrict__ dinv) {
    const int i = blockIdx.x * blockDim.x + threadIdx.x;
    if (i < NTOT) {
        const float d = deg[i];
        dinv[i] = (d > 0.f) ? rsqrtf(d) : 0.f;
    }
}

// ---------------------------------------------------------------------------
// Kernel: scatter  out[dst] += xw[src] * dinv[src]*dinv[dst]  (256 ch/edge)
// ---------------------------------------------------------------------------
__global__ void __launch_bounds__(256)
gcn_scatter_kernel(const int* __restrict__ src, const int* __restrict__ dst,
                   const float* __restrict__ xw, const float* __restrict__ dinv,
                   float* __restrict__ outg)
{
    const int e = blockIdx.x;
    const int s = src[e];
    const int d = dst[e];
    const float nrm = dinv[s] * dinv[d];
    const int c = threadIdx.x;
    atomicAdd(&outg[(size_t)d * C_ + c], xw[(size_t)s * C_ + c] * nrm);
}

// ---------------------------------------------------------------------------
// Kernel: x = relu(outg + xw*dinv^2 (self loop) + b_gcn) + state
// ---------------------------------------------------------------------------
__global__ void __launch_bounds__(256)
gcn_combine_kernel(const float* __restrict__ outg, const float* __restrict__ xw,
                   const float* __restrict__ dinv, const float* __restrict__ bg,
                   const float* __restrict__ state, float* __restrict__ x)
{
    const size_t i = (size_t)blockIdx.x * blockDim.x + threadIdx.x;
    if (i >= (size_t)NTOT * C_) return;
    const int node = (int)(i >> 8);
    const int c    = (int)(i & 255);
    const float di = dinv[node];
    float v = outg[i] + xw[i] * di * di + bg[c];
    v = (v > 0.f) ? v : 0.f;
    x[i] = v + state[i];
}

// ---------------------------------------------------------------------------
// Fused MLP kernel: one workgroup per (batch b, 16-edge strip).
//   TDM gather-mode DMAs pull the 16 src rows and 16 dst rows of x[b] into
//   LDS (row indices packed into D# groups 2/3), action column loaded by ALU.
//   Layer1 = two K=256 WMMA passes (src half, dst half) + rank-1 action term.
//   Layer2 WMMA from LDS, leaky_relu, masked column sum -> atomicAdd hs[b,:].
// grid = (ceil(E/16)=63, B), 256 threads (8 waves x 2 N-tiles)
// ---------------------------------------------------------------------------
#define ETILES ((E_ + 15) / 16)   // 63

__global__ void __launch_bounds__(256)
mlp_fused_kernel(const float* __restrict__ x, const float* __restrict__ action,
                 const int* __restrict__ edges,
                 const float* __restrict__ W1, const float* __restrict__ b1,
                 const float* __restrict__ W2, const float* __restrict__ b2,
                 float* __restrict__ hs)
{
    __shared__ float Asrc[16 * LDAP];   // 16.9 KB
    __shared__ float Adst[16 * LDAP];   // 16.9 KB
    __shared__ float Aact[16];
    __shared__ float H1[16 * LDAP];     // 16.9 KB

    const int b  = blockIdx.y;
    const int e0 = blockIdx.x * 16;

    if (threadIdx.x < 16) {
        int e = e0 + threadIdx.x; if (e >= E_) e = E_ - 1;
        Aact[threadIdx.x] = action[(size_t)b * E_ + e];
    }

#if USE_TDM
    if (threadIdx.x < 32) {   // wave 0 issues both gather DMAs
        int si[16], di[16];
        #pragma unroll
        for (int r = 0; r < 16; ++r) {
            int e = e0 + r; if (e >= E_) e = E_ - 1;
            si[r] = __builtin_amdgcn_readfirstlane(edges[2 * e]);
            di[r] = __builtin_amdgcn_readfirstlane(edges[2 * e + 1]);
        }
        i32x4 s2, s3, d2, d3;
        #pragma unroll
        for (int j = 0; j < 4; ++j) {
            s2[j] = (si[2*j]     & 0xffff) | (si[2*j + 1]     << 16);
            s3[j] = (si[8 + 2*j] & 0xffff) | (si[8 + 2*j + 1] << 16);
            d2[j] = (di[2*j]     & 0xffff) | (di[2*j + 1]     << 16);
            d3[j] = (di[8 + 2*j] & 0xffff) | (di[8 + 2*j + 1] << 16);
        }
        const unsigned long long base =
            (unsigned long long)(uintptr_t)(x + (size_t)b * N_ * C_);
        tdm_load_to_lds(lds_off_of(Asrc), base,
                        /*td0=*/C_, /*td1=*/N_, /*tile0=*/C_, /*nIdx=*/16,
                        /*stride0=*/C_, /*gather=*/true, s2, s3);
        tdm_load_to_lds(lds_off_of(Adst), base,
                        /*td0=*/C_, /*td1=*/N_, /*tile0=*/C_, /*nIdx=*/16,
                        /*stride0=*/C_, /*gather=*/true, d2, d3);
        __builtin_amdgcn_s_wait_tensorcnt(0);
    }
#else
    for (int idx = threadIdx.x; idx < 16 * 128; idx += blockDim.x) {
        const int r = idx >> 7;
        const int q = idx & 127;         // 0..63 src half, 64..127 dst half
        int e = e0 + r; if (e >= E_) e = E_ - 1;
        const int node = (q < 64) ? edges[2 * e] : edges[2 * e + 1];
        const int c4   = (q & 63) * 4;
        const float4 v = *reinterpret_cast<const float4*>(
            x + ((size_t)b * N_ + node) * C_ + c4);
        float* dp = (q < 64) ? &Asrc[r * LDAP + c4] : &Adst[r * LDAP + c4];
        *reinterpret_cast<float4*>(dp) = v;
    }
#endif
    __syncthreads();

    const int wave = threadIdx.x >> 5;
    const int lane = threadIdx.x & 31;
    const int mhi  = (lane >> 4) * 8;

    // ---- layer 1: h1 = leaky_relu([src|dst|act] @ W1 + b1) --------------
    #pragma unroll
    for (int t = 0; t < 2; ++t) {
        const int n0 = (wave * 2 + t) * 16;
        const int n  = n0 + (lane & 15);
        v8f acc = {0.f,0.f,0.f,0.f,0.f,0.f,0.f,0.f};
        acc = wmma_tile_f32(Asrc, LDAP, W1,                    H_, n0, C_, acc);
        acc = wmma_tile_f32(Adst, LDAP, W1 + (size_t)C_ * H_,  H_, n0, C_, acc);
        const float w512 = W1[(size_t)(2 * C_) * H_ + n];      // action column
        const float bias = b1[n];
        #pragma unroll
        for (int v = 0; v < 8; ++v) {
            const int m = mhi + v;
            float val = acc[v] + Aact[m] * w512 + bias;
            H1[m * LDAP + n] = (val > 0.f) ? val : 0.01f * val;
        }
    }
    __syncthreads();

    // ---- layer 2 + masked column sum ------------------------------------
    #pragma unroll
    for (int t = 0; t < 2; ++t) {
        const int n0 = (wave * 2 + t) * 16;
        const int n  = n0 + (lane & 15);
        v8f acc = {0.f,0.f,0.f,0.f,0.f,0.f,0.f,0.f};
        acc = wmma_tile_f32(H1, LDAP, W2, H_, n0, H_, acc);
        const float bias = b2[n];
        float sum = 0.f;
        #pragma unroll
        for (int v = 0; v < 8; ++v) {
            const int m = mhi + v;
            float val = acc[v] + bias;
            val = (val > 0.f) ? val : 0.01f * val;
            if (e0 + m < E_) sum += val;     // mask padded rows (E=1000)
        }
        sum += __shfl_xor(sum, 16, 32);      // combine the two row-halves
        if (lane < 16) atomicAdd(&hs[(size_t)b * H_ + n], sum);
    }
}

// ---------------------------------------------------------------------------
// Final: out[b] = hs[b,:] @ W3 + b3
// ---------------------------------------------------------------------------
__global__ void __launch_bounds__(256)
final_out_kernel(const float* __restrict__ hs, const float* __restrict__ W3,
                 const float* __restrict__ b3, float* __restrict__ out)
{
    __shared__ float red[8];
    const int b = blockIdx.x;
    float v = hs[(size_t)b * H_ + threadIdx.x] * W3[threadIdx.x];
    #pragma unroll
    for (int o = 16; o > 0; o >>= 1) v += __shfl_down(v, o, 32);
    if ((threadIdx.x & 31) == 0) red[threadIdx.x >> 5] = v;
    __syncthreads();
    if (threadIdx.x == 0) {
        float s = 0.f;
        #pragma unroll
        for (int i = 0; i < 8; ++i) s += red[i];
        out[b] = s + b3[0];
    }
}

// ---------------------------------------------------------------------------
// Host launcher
// ---------------------------------------------------------------------------
extern "C" void kernel_launch(void* const* d_in, const int* in_sizes, int n_in,
                              void* d_out, int out_size, void* d_ws, size_t ws_size,
                              hipStream_t stream) {
    const float* state  = (const float*)d_in[0];
    const float* action = (const float*)d_in[1];
    const float* W_gcn  = (const float*)d_in[2];
    const float* b_gcn  = (const float*)d_in[3];
    const float* W1     = (const float*)d_in[4];
    const float* b1     = (const float*)d_in[5];
    const float* W2     = (const float*)d_in[6];
    const float* b2     = (const float*)d_in[7];
    const float* W3     = (const float*)d_in[8];
    const float* b3     = (const float*)d_in[9];
    const int*   eidx   = (const int*)d_in[10];   // (2, EG): row0=src, row1=dst
    const int*   edges  = (const int*)d_in[11];   // (E, 2)
    float* out = (float*)d_out;

    // Workspace layout (floats)
    float* ws    = (float*)d_ws;
    float* xw    = ws;                                   // NTOT*C
    float* outg  = xw   + (size_t)NTOT * C_;             // NTOT*C
    float* xbuf  = outg + (size_t)NTOT * C_;             // NTOT*C
    float* deg   = xbuf + (size_t)NTOT * C_;             // NTOT
    float* dinv  = deg  + NTOT;                          // NTOT
    float* hs    = dinv + NTOT;                          // B*H

    const int* e_src = eidx;
    const int* e_dst = eidx + EG_;

    // 1) xw = state @ W_gcn
    gemm_xw_kernel<<<NTOT / 16, 256, 0, stream>>>(state, W_gcn, xw);

    // 2) degrees and normalization
    deg_init_kernel<<<(NTOT + 255) / 256, 256, 0, stream>>>(deg);
    deg_count_kernel<<<(EG_ + 255) / 256, 256, 0, stream>>>(e_dst, deg);
    deg_rsqrt_kernel<<<(NTOT + 255) / 256, 256, 0, stream>>>(deg, dinv);

    // 3) scatter-accumulate
    hipMemsetAsync(outg, 0, (size_t)NTOT * C_ * sizeof(float), stream);
    gcn_scatter_kernel<<<EG_, 256, 0, stream>>>(e_src, e_dst, xw, dinv, outg);

    // 4) x = relu(out + self + b) + state
    gcn_combine_kernel<<<(NTOT * C_) / 256, 256, 0, stream>>>(
        outg, xw, dinv, b_gcn, state, xbuf);

    // 5) fused two-layer MLP + sum over E
    hipMemsetAsync(hs, 0, (size_t)B_ * H_ * sizeof(float), stream);
    mlp_fused_kernel<<<dim3(ETILES, B_), 256, 0, stream>>>(
        xbuf, action, edges, W1, b1, W2, b2, hs);

    // 6) out = hs @ W3 + b3
    final_out_kernel<<<B_, 256, 0, stream>>>(hs, W3, b3, out);
}